// CDMF_7919919694076
// MI455X (gfx1250) — compile-verified
//
#include <hip/hip_runtime.h>
#include <hip/hip_bf16.h>
#include <cstdint>
#include <cstddef>

// ---------------------------------------------------------------------------
// Problem constants (match reference)
// ---------------------------------------------------------------------------
#define N_USERS_C   10000
#define N_ITEMS_C   100000
#define NFEAT       64
#define EMB         128
#define NSEQ        100000
#define SLEN        50
#define TAU_C       0.01f

static constexpr int    NROWS  = NSEQ * SLEN;   // 5,000,000 rows of R (flattened n,s)
static constexpr int    NTILES = NROWS / 16;    // 312,500 exact 16-row WMMA tiles

typedef __attribute__((ext_vector_type(2))) float v2f;
typedef __attribute__((ext_vector_type(4))) float v4f;
typedef __attribute__((ext_vector_type(8))) float v8f;

// ---------------------------------------------------------------------------
// Kernel A: Z[r] = sum_k R[r,k] * w[k]  for r in [0, 5M), K = 64
// Mapped to V_WMMA_F32_16X16X4_F32: A = 16x4 tile of R, B[k][n] = w[kbase+k]
// (broadcast over N => every column of D holds Z for the 16 rows).
// f32 in / f32 accumulate => numerically identical to the reference einsum.
// ---------------------------------------------------------------------------
__global__ void zcalc_wmma(const float* __restrict__ R,
                           const float* __restrict__ w,
                           float* __restrict__ Z) {
  const int lane = threadIdx.x & 31;
  const int wid  = threadIdx.x >> 5;
  const int wavesPerBlock = blockDim.x >> 5;
  const int gwave  = blockIdx.x * wavesPerBlock + wid;
  const int nwaves = gridDim.x * wavesPerBlock;

  const int half = lane >> 4;    // 0: lanes 0-15 carry K = base,base+1 ; 1: K = base+2,base+3
  const int mrow = lane & 15;    // A-matrix row within the tile

  // B operand (invariant across tiles): B[k][n] = w[4j + k] broadcast over n.
  // f32 4x16 B layout: VGPR0/1 x {low lanes K=0,1 | high lanes K=2,3}.
  v2f b[16];
#pragma unroll
  for (int j = 0; j < 16; ++j) {
    const int k = 4 * j + 2 * half;
    b[j].x = w[k];
    b[j].y = w[k + 1];
  }

  for (int tile = gwave; tile < NTILES; tile += nwaves) {
    // A operand: lane(half=0) reads R[row, 4j..4j+1], lane(half=1) reads R[row, 4j+2..4j+3].
    // The wave's 16 float2 loads collectively consume the contiguous 4 KB tile.
    const float* ap = R + (size_t)(tile * 16 + mrow) * NFEAT + 2 * half;
    v2f a[16];
#pragma unroll
    for (int j = 0; j < 16; ++j)
      a[j] = *(const v2f*)(ap + 4 * j);

    v8f c = {};  // zero accumulator
#pragma unroll
    for (int j = 0; j < 16; ++j)
      c = __builtin_amdgcn_wmma_f32_16x16x4_f32(
              /*neg_a=*/false, a[j], /*neg_b=*/false, b[j],
              /*c_mod=*/(short)0, c, /*reuse_a=*/false, /*reuse_b=*/false);

    // D column 0 lives in lanes 0 (M=0..7) and 16 (M=8..15); write Z from those lanes.
    if (mrow == 0) {
      float* zp = Z + tile * 16 + 8 * half;
      *(v4f*)(zp)     = v4f{c[0], c[1], c[2], c[3]};
      *(v4f*)(zp + 4) = v4f{c[4], c[5], c[6], c[7]};
    }
  }
}

// ---------------------------------------------------------------------------
// Kernel B: wt[n] = ((sum_s mask*max(Z,tau)^alpha)^(1/alpha) * cnt^beta)^gamma
// One wave32 per sequence; lanes cover s = lane, lane+32.
// ---------------------------------------------------------------------------
__global__ void wt_calc(const float* __restrict__ Z,
                        const unsigned char* __restrict__ mask,
                        const float* __restrict__ pAlpha,
                        const float* __restrict__ pBeta,
                        const float* __restrict__ pGamma,
                        float* __restrict__ wt) {
  const int lane = threadIdx.x & 31;
  const int n = blockIdx.x * (blockDim.x >> 5) + (threadIdx.x >> 5);
  if (n >= NSEQ) return;  // wave-uniform exit

  const float alpha = pAlpha[0];
  const float beta  = pBeta[0];
  const float gamma = pGamma[0];

  float sum = 0.0f, cnt = 0.0f;
  for (int s = lane; s < SLEN; s += 32) {
    const size_t idx = (size_t)n * SLEN + s;
    const bool  m  = mask[idx] != 0;
    const float wv = fmaxf(Z[idx], TAU_C);
    if (m) { sum += __powf(wv, alpha); cnt += 1.0f; }
  }
#pragma unroll
  for (int off = 16; off > 0; off >>= 1) {
    sum += __shfl_xor(sum, off, 32);
    cnt += __shfl_xor(cnt, off, 32);
  }
  if (lane == 0) {
    const float w_alpha = __powf(sum, 1.0f / alpha);
    const float r_beta  = __powf(cnt, beta);
    wt[n] = __powf(w_alpha * r_beta, gamma);
  }
}

// ---------------------------------------------------------------------------
// Kernel C: segment sums  num[u,:] += wt[n]*q[n,:],  den[u] += wt[n]
// One 128-thread block per sequence; global_atomic_add_f32.
// ---------------------------------------------------------------------------
__global__ void segsum(const int* __restrict__ users,
                       const int* __restrict__ items,
                       const float* __restrict__ emb,
                       const float* __restrict__ wt,
                       float* __restrict__ num,
                       float* __restrict__ den) {
  const int n = blockIdx.x;
  const int d = threadIdx.x;  // 0..127
  const int u  = users[n];
  const int it = items[n];
  const float wn = wt[n];
  const float q  = emb[(size_t)it * EMB + d];
  atomicAdd(&num[(size_t)u * EMB + d], wn * q);
  if (d == 0) atomicAdd(&den[u], wn);
}

// ---------------------------------------------------------------------------
// Kernel D: r[n] = (sum_d num[u,d]*q[n,d]) / den[u]
// One wave32 per sequence; 4 dims per lane, shuffle reduction.
// ---------------------------------------------------------------------------
__global__ void output_r(const int* __restrict__ users,
                         const int* __restrict__ items,
                         const float* __restrict__ emb,
                         const float* __restrict__ num,
                         const float* __restrict__ den,
                         float* __restrict__ r) {
  const int lane = threadIdx.x & 31;
  const int n = blockIdx.x * (blockDim.x >> 5) + (threadIdx.x >> 5);
  if (n >= NSEQ) return;  // wave-uniform exit

  const int u  = users[n];
  const int it = items[n];
  float acc = 0.0f;
#pragma unroll
  for (int k = 0; k < 4; ++k) {
    const int d = lane + 32 * k;
    acc += num[(size_t)u * EMB + d] * emb[(size_t)it * EMB + d];
  }
#pragma unroll
  for (int off = 16; off > 0; off >>= 1)
    acc += __shfl_xor(acc, off, 32);
  if (lane == 0) r[n] = acc / den[u];
}

// ---------------------------------------------------------------------------
// Launcher
// ---------------------------------------------------------------------------
extern "C" void kernel_launch(void* const* d_in, const int* in_sizes, int n_in,
                              void* d_out, int out_size, void* d_ws, size_t ws_size,
                              hipStream_t stream) {
  (void)in_sizes; (void)n_in; (void)out_size; (void)ws_size;

  const int*           users = (const int*)d_in[0];
  const int*           items = (const int*)d_in[1];
  const float*         R     = (const float*)d_in[2];
  const unsigned char* mask  = (const unsigned char*)d_in[3];  // jax bool = 1 byte
  const float*         w     = (const float*)d_in[4];
  const float*         emb   = (const float*)d_in[5];
  const float*         alpha = (const float*)d_in[6];
  const float*         beta  = (const float*)d_in[7];
  const float*         gamma = (const float*)d_in[8];
  float*               out   = (float*)d_out;

  // Workspace layout (256B-aligned regions):
  //   Z   : 5,000,000 f32  (20.0 MB)
  //   wt  :   100,000 f32  ( 0.4 MB)
  //   num : 10001*128 f32  ( 5.1 MB)   } contiguous, zeroed each call
  //   den : 10001     f32  ( 40 KB)    }
  auto align256 = [](size_t x) { return (x + 255) & ~(size_t)255; };
  char*  ws     = (char*)d_ws;
  size_t offZ   = 0;
  size_t offWt  = align256(offZ  + (size_t)NROWS * sizeof(float));
  size_t offNum = align256(offWt + (size_t)NSEQ  * sizeof(float));
  size_t offDen = offNum + (size_t)(N_USERS_C + 1) * EMB * sizeof(float);
  size_t endOff = offDen + (size_t)(N_USERS_C + 1) * sizeof(float);

  float* Z   = (float*)(ws + offZ);
  float* wt  = (float*)(ws + offWt);
  float* num = (float*)(ws + offNum);
  float* den = (float*)(ws + offDen);

  // Zero the accumulation buffers every call (deterministic, capture-safe).
  hipMemsetAsync(ws + offNum, 0, endOff - offNum, stream);

  // A: WMMA einsum. 4096 blocks x 8 waves => 32768 waves grid-striding 312500 tiles.
  zcalc_wmma<<<4096, 256, 0, stream>>>(R, w, Z);

  // B: per-sequence weight. 8 waves/block.
  wt_calc<<<(NSEQ + 7) / 8, 256, 0, stream>>>(Z, mask, alpha, beta, gamma, wt);

  // C: per-user segment sums via f32 atomics.
  segsum<<<NSEQ, EMB, 0, stream>>>(users, items, emb, wt, num, den);

  // D: final per-sequence dot.
  output_r<<<(NSEQ + 7) / 8, 256, 0, stream>>>(users, items, emb, num, den, out);
}